// QuantLiftSplatShoot_15899968930296
// MI455X (gfx1250) — compile-verified
//
#include <hip/hip_runtime.h>
#include <hip/hip_bf16.h>

// ---------------- LSS constants (from reference) ----------------
#define BB   4
#define NN   4
#define DD   41
#define FH   30
#define FW   40
#define CC   64
#define FHW  (FH*FW)            // 1200
#define PPB  (DD*FHW)           // 49200 points per (b,n)
#define GRP  (PPB/16)           // 3075 WMMA groups of 16 points per (b,n)
#define NXY  65536              // 256*256 BEV cells per batch

typedef __attribute__((ext_vector_type(2))) float v2f;
typedef __attribute__((ext_vector_type(8))) float v8f;

// ---------------- zero-fill output: b128 stores ----------------
__global__ void lss_zero(float4* __restrict__ out) {
    size_t i = (size_t)blockIdx.x * blockDim.x + threadIdx.x;
    out[i] = make_float4(0.f, 0.f, 0.f, 0.f);
}

// ---------------- 3x3 inverse (adjugate / det) ----------------
__device__ inline void inv3(const float* m, float* r) {
    float a = m[0], b = m[1], c = m[2];
    float d = m[3], e = m[4], f = m[5];
    float g = m[6], h = m[7], i = m[8];
    float A =  (e*i - f*h), B = -(d*i - f*g), C =  (d*h - e*g);
    float det = a*A + b*B + c*C;
    float id = 1.0f / det;
    r[0] = A * id;            r[1] = -(b*i - c*h) * id; r[2] =  (b*f - c*e) * id;
    r[3] = B * id;            r[4] =  (a*i - c*g) * id; r[5] = -(a*f - c*d) * id;
    r[6] = C * id;            r[7] = -(a*h - b*g) * id; r[8] =  (a*e - b*d) * id;
}

// ---------------- per-(b,n) matrices into workspace ----------------
// layout per bn (24 floats): combine[3][3], trans[3], invPostRot[3][3], post_trans[3]
__global__ void lss_prep(const float* __restrict__ rots,
                         const float* __restrict__ trans,
                         const float* __restrict__ intrins,
                         const float* __restrict__ post_rots,
                         const float* __restrict__ post_trans,
                         float* __restrict__ mats) {
    int bn = threadIdx.x;
    if (bn >= BB * NN) return;
    float ik[9];
    inv3(intrins + bn * 9, ik);
    const float* R = rots + bn * 9;
    float* M = mats + bn * 24;
    #pragma unroll
    for (int r = 0; r < 3; ++r)
        #pragma unroll
        for (int c = 0; c < 3; ++c)
            M[r*3 + c] = R[r*3+0]*ik[0*3+c] + R[r*3+1]*ik[1*3+c] + R[r*3+2]*ik[2*3+c];
    #pragma unroll
    for (int k = 0; k < 3; ++k) M[9 + k] = trans[bn*3 + k];
    inv3(post_rots + bn * 9, M + 12);
    #pragma unroll
    for (int k = 0; k < 3; ++k) M[21 + k] = post_trans[bn*3 + k];
}

// ---------------- geometry via V_WMMA_F32_16X16X4_F32 ----------------
// 16 points per wave per iteration:
//   D(16x16) = A(16x4) x B(4x16),  A rows 0..2 = [combine | trans],
//   B columns = (x*z, y*z, z, 1) per point.  D rows 0..2 = geom xyz.
// f32 A layout: lanes 0-15 hold K=0(v0),K=1(v1); lanes 16-31 hold K=2(v0),K=3(v1).
// Both lane halves recompute the same point's stage-1 so no shuffles are needed.
__global__ void lss_geom(const float* __restrict__ frustum,
                         const float* __restrict__ mats,
                         const float* __restrict__ dx,
                         const float* __restrict__ bx,
                         int* __restrict__ vidx) {
    const int bn   = blockIdx.y;            // 0..15
    const int lane = threadIdx.x & 31;
    const int wv   = threadIdx.x >> 5;
    const int sub  = lane & 15;
    const int half = lane >> 4;
    const int b    = bn >> 2;               // bn / NN

    const float* M   = mats + bn * 24;
    const float* IPR = M + 12;
    const float pt0 = M[21], pt1 = M[22], pt2 = M[23];

    // A matrix (constant across the loop)
    v2f Av; Av.x = 0.f; Av.y = 0.f;
    if (sub < 3) {
        if (half == 0) { Av.x = M[sub*3 + 0]; Av.y = M[sub*3 + 1]; }
        else           { Av.x = M[sub*3 + 2]; Av.y = M[9 + sub];   }
    }

    const float dx0 = dx[0], dx1 = dx[1], dx2 = dx[2];
    const float ox = bx[0] - 0.5f*dx0, oy = bx[1] - 0.5f*dx1, oz = bx[2] - 0.5f*dx2;
    const float rdx0 = 1.0f/dx0, rdx1 = 1.0f/dx1, rdx2 = 1.0f/dx2;

    const int wave = blockIdx.x * (blockDim.x >> 5) + wv;
    const int nw   = gridDim.x * (blockDim.x >> 5);

    for (int g = wave; g < GRP; g += nw) {      // uniform per wave -> EXEC all 1s at WMMA
        const int p = g * 16 + sub;             // point index within (b,n): (d,h,w) row-major
        const float* fr = frustum + p * 3;      // frustum is [D,FH,FW,3] == p*3
        const float px = fr[0] - pt0;
        const float py = fr[1] - pt1;
        const float pz = fr[2] - pt2;
        const float qx = IPR[0]*px + IPR[1]*py + IPR[2]*pz;
        const float qy = IPR[3]*px + IPR[4]*py + IPR[5]*pz;
        const float qz = IPR[6]*px + IPR[7]*py + IPR[8]*pz;

        v2f Bv;
        if (half == 0) { Bv.x = qx * qz; Bv.y = qy * qz; }   // K=0,1 rows
        else           { Bv.x = qz;      Bv.y = 1.0f;    }   // K=2,3 rows
        v8f Cv = {};
        v8f Dv = __builtin_amdgcn_wmma_f32_16x16x4_f32(
                     false, Av, false, Bv, (short)0, Cv, false, false);

        if (half == 0) {                         // lanes 0-15 hold rows 0..7 -> geom xyz
            const int ix = (int)floorf((Dv[0] - ox) * rdx0);
            const int iy = (int)floorf((Dv[1] - oy) * rdx1);
            const int iz = (int)floorf((Dv[2] - oz) * rdx2);
            int vox = -1;
            if (ix >= 0 && ix < 256 && iy >= 0 && iy < 256 && iz == 0)
                vox = (b << 16) | (ix << 8) | iy;
            vidx[bn * PPB + p] = vox;
        }
    }
}

// ---------------- splat: softmax + outer product + scatter atomics ----------------
// one wave per pixel; lanes cover channels {lane, lane+32} -> 2 atomics per depth bin.
__global__ void lss_splat(const float* __restrict__ feat,
                          const float* __restrict__ dlog,
                          const int*   __restrict__ vidx,
                          float* __restrict__ out) {
    const int wv   = threadIdx.x >> 5;
    const int lane = threadIdx.x & 31;
    const int pix  = blockIdx.x * 8 + wv;       // exactly 19200 pixels, 2400 blocks
    const int bn   = pix / FHW;
    const int hw   = pix - bn * FHW;
    const int b    = bn >> 2;

    const float* dlp = dlog + (size_t)bn * DD * FHW + hw;
    const float* fp  = feat + (size_t)bn * CC * FHW + hw;
    __builtin_prefetch(fp + (size_t)lane * FHW, 0, 0);          // global_prefetch_b8
    __builtin_prefetch(fp + (size_t)(lane + 32) * FHW, 0, 0);

    // softmax over 41 depth bins (wave32 reductions)
    float l0 = dlp[lane * FHW];
    float l1 = (lane < DD - 32) ? dlp[(32 + lane) * FHW] : -3.4e38f;
    float m = fmaxf(l0, l1);
    #pragma unroll
    for (int s = 16; s > 0; s >>= 1) m = fmaxf(m, __shfl_xor(m, s, 32));
    float e0 = __expf(l0 - m);
    float e1 = (lane < DD - 32) ? __expf(l1 - m) : 0.f;
    float sum = e0 + e1;
    #pragma unroll
    for (int s = 16; s > 0; s >>= 1) sum += __shfl_xor(sum, s, 32);
    const float inv = 1.0f / sum;

    __shared__ float wlds[8][48];
    __shared__ int   vlds[8][48];
    wlds[wv][lane] = e0 * inv;
    const int* vp = vidx + bn * PPB + hw;
    vlds[wv][lane] = vp[lane * FHW];
    if (lane < DD - 32) {
        wlds[wv][32 + lane] = e1 * inv;
        vlds[wv][32 + lane] = vp[(32 + lane) * FHW];
    }
    // same-wave LDS writes/reads are in-order; no block barrier needed.

    const float f0 = fp[lane * FHW];
    const float f1 = fp[(32 + lane) * FHW];

    float* ob = out + ((size_t)(b * CC) << 16);       // out[b][c][x][y], c stride = 65536
    #pragma unroll 1
    for (int d = 0; d < DD; ++d) {
        const int   vox = vlds[wv][d];
        const float wd  = wlds[wv][d];
        if (vox >= 0) {                               // uniform across the wave
            float* o = ob + (vox & 0xFFFF);
            atomicAdd(o + ((size_t)lane << 16),        wd * f0);
            atomicAdd(o + ((size_t)(lane + 32) << 16), wd * f1);
        }
    }
}

// ---------------- launch ----------------
extern "C" void kernel_launch(void* const* d_in, const int* in_sizes, int n_in,
                              void* d_out, int out_size, void* d_ws, size_t ws_size,
                              hipStream_t stream) {
    const float* feat       = (const float*)d_in[0];
    const float* depth_log  = (const float*)d_in[1];
    const float* frustum    = (const float*)d_in[2];
    const float* rots       = (const float*)d_in[3];
    const float* trans      = (const float*)d_in[4];
    const float* intrins    = (const float*)d_in[5];
    const float* post_rots  = (const float*)d_in[6];
    const float* post_trans = (const float*)d_in[7];
    const float* dx         = (const float*)d_in[8];
    const float* bx         = (const float*)d_in[9];
    float* out = (float*)d_out;

    float* mats = (float*)d_ws;                       // 16*24 floats
    int*   vidx = (int*)((char*)d_ws + 2048);         // 787200 ints

    // out: 4*64*65536 floats = 4,194,304 float4 -> 16384 blocks x 256
    lss_zero<<<16384, 256, 0, stream>>>((float4*)out);
    lss_prep<<<1, 32, 0, stream>>>(rots, trans, intrins, post_rots, post_trans, mats);
    lss_geom<<<dim3(16, BB * NN), 256, 0, stream>>>(frustum, mats, dx, bx, vidx);
    lss_splat<<<(BB * NN * FHW) / 8, 256, 0, stream>>>(feat, depth_log, vidx, out);
}